// ResBlock1d_F_14431090114935
// MI455X (gfx1250) — compile-verified
//
#include <hip/hip_runtime.h>
#include <hip/hip_bf16.h>

typedef __attribute__((ext_vector_type(2))) float v2f;
typedef __attribute__((ext_vector_type(8))) float v8f;

namespace {
constexpr int B_N  = 32;
constexpr int CIN  = 64;
constexpr int CMID = 128;
constexpr int COUT = 256;
constexpr int LIN  = 8192;
constexpr int L2   = 4096;
constexpr int WTOT = 512;               // OUT_N + 2*MID_N rows in all_W/all_b
constexpr float EPSV = 1e-5f;
constexpr float INVN = 1.0f / (32.0f * 4096.0f);   // 1/(B*L2)

// workspace layout (float offsets)
constexpr size_t P2S = 0;               // [128][32] bn2 sum partials
constexpr size_t P2Q = 4096;            // [128][32] bn2 sumsq partials
constexpr size_t P3S = 8192;            // [256][32]
constexpr size_t P3Q = 16384;           // [256][32]
constexpr size_t F2S = 24576;           // [128] final bn2 sums
constexpr size_t F2Q = 24704;
constexpr size_t F3S = 24832;           // [256]
constexpr size_t F3Q = 25088;
constexpr size_t H2O = 32768;           // h2: [32][128][4096]
constexpr size_t OUT_ELEMS = (size_t)B_N * COUT * L2;
} // namespace

// ---- kernel 0: write the tuple's integer output (i = from_idx + 512) ----
__global__ void k0_tail(float* __restrict__ out, const int* __restrict__ fip,
                        int out_size) {
  if (threadIdx.x == 0 && blockIdx.x == 0) {
    if ((size_t)out_size > OUT_ELEMS)
      out[OUT_ELEMS] = (float)(fip[0] + 2 * CMID + COUT);
  }
}

// ---- kernel 1: h2 = dynconv2(relu(dynconv1(x)+b1), stride 2); bn2 partials ----
// One block per (b, x-channel j); thread = 16 consecutive t outputs for both
// output channels c=2j, 2j+1 (they share the same x row -> x read once).
__global__ void __launch_bounds__(256)
k1_conv12(const float* __restrict__ x, const float* __restrict__ allW,
          const float* __restrict__ allb, const int* __restrict__ fip,
          float* __restrict__ ws) {
  const int fi = fip[0];
  const int b = blockIdx.x >> 6;
  const int j = blockIdx.x & 63;
  const int c0 = 2 * j;
  const float* __restrict__ xr = x + ((size_t)b * CIN + j) * LIN;

  const float* W1 = allW + ((size_t)b * WTOT + fi + c0) * 3;
  const float* W2 = allW + ((size_t)b * WTOT + fi + CMID + c0) * 3;
  const float w10a = W1[0], w11a = W1[1], w12a = W1[2];
  const float w10b = W1[3], w11b = W1[4], w12b = W1[5];
  const float w20a = W2[0], w21a = W2[1], w22a = W2[2];
  const float w20b = W2[3], w21b = W2[4], w22b = W2[5];
  const float b1a = allb[(size_t)b * WTOT + fi + c0];
  const float b1b = allb[(size_t)b * WTOT + fi + c0 + 1];

  float* __restrict__ h2a = ws + H2O + ((size_t)b * CMID + c0) * L2;
  float* __restrict__ h2b = h2a + L2;

  const int t0 = threadIdx.x * 16;
  const int s0 = 2 * t0;
  // sliding x window: xB = X(2t-1), xC = X(2t); pad handled by clamp+select
  const float xBl = xr[max(s0 - 1, 0)];
  float xB = (s0 >= 1) ? xBl : 0.0f;
  float xC = xr[s0];
  const float xAl = xr[max(s0 - 2, 0)];
  float h1ma = fmaxf(b1a + w10a * xAl + w11a * xB + w12a * xC, 0.0f);
  float h1mb = fmaxf(b1b + w10b * xAl + w11b * xB + w12b * xC, 0.0f);
  if (t0 == 0) { h1ma = 0.0f; h1mb = 0.0f; }   // h1(-1) is conv padding -> 0

  float sa = 0.f, qa = 0.f, sb = 0.f, qb = 0.f;
#pragma unroll 4
  for (int t = t0; t < t0 + 16; ++t) {
    const int sD = 2 * t + 1;                    // <= 8191 always
    const float xD = xr[sD];
    const float xEl = xr[min(sD + 1, LIN - 1)];
    const float xE = (sD + 1 < LIN) ? xEl : 0.0f;
    const float h1ca = fmaxf(b1a + w10a * xB + w11a * xC + w12a * xD, 0.0f);
    const float h1pa = fmaxf(b1a + w10a * xC + w11a * xD + w12a * xE, 0.0f);
    const float h1cb = fmaxf(b1b + w10b * xB + w11b * xC + w12b * xD, 0.0f);
    const float h1pb = fmaxf(b1b + w10b * xC + w11b * xD + w12b * xE, 0.0f);
    const float va = w20a * h1ma + w21a * h1ca + w22a * h1pa;
    const float vb = w20b * h1mb + w21b * h1cb + w22b * h1pb;
    h2a[t] = va; h2b[t] = vb;
    sa += va; qa += va * va; sb += vb; qb += vb * vb;
    h1ma = h1pa; h1mb = h1pb; xB = xD; xC = xE;
  }

  __shared__ float4 red[256];
  red[threadIdx.x] = make_float4(sa, qa, sb, qb);
  __syncthreads();
  for (int off = 128; off > 0; off >>= 1) {
    if (threadIdx.x < off) {
      float4 u = red[threadIdx.x], v = red[threadIdx.x + off];
      red[threadIdx.x] = make_float4(u.x + v.x, u.y + v.y, u.z + v.z, u.w + v.w);
    }
    __syncthreads();
  }
  if (threadIdx.x == 0) {
    float4 r = red[0];
    ws[P2S + (size_t)c0 * 32 + b] = r.x;
    ws[P2Q + (size_t)c0 * 32 + b] = r.y;
    ws[P2S + (size_t)(c0 + 1) * 32 + b] = r.z;
    ws[P2Q + (size_t)(c0 + 1) * 32 + b] = r.w;
  }
}

// ---- deterministic fold of the 32 per-batch partials per channel ----
__global__ void k_red32(const float* __restrict__ ps, const float* __restrict__ pq,
                        float* __restrict__ fs, float* __restrict__ fq, int nch) {
  const int ch = blockIdx.x * blockDim.x + threadIdx.x;
  if (ch >= nch) return;
  float s = 0.f, q = 0.f;
#pragma unroll
  for (int bb = 0; bb < 32; ++bb) {
    s += ps[(size_t)ch * 32 + bb];
    q += pq[(size_t)ch * 32 + bb];
  }
  fs[ch] = s; fq[ch] = q;
}

// ---- kernel 2: bn3 statistics of h3 = dynconv3(relu(bn2(h2))) (h3 NOT stored) ----
__global__ void __launch_bounds__(256)
k2_conv3_stats(const float* __restrict__ allW, const float* __restrict__ ln2w,
               const float* __restrict__ ln2b, const int* __restrict__ fip,
               float* __restrict__ ws) {
  const int fi = fip[0];
  const int b = blockIdx.x >> 7;
  const int j = blockIdx.x & 127;
  const float* __restrict__ h2r = ws + H2O + ((size_t)b * CMID + j) * L2;
  const float m   = ws[F2S + j] * INVN;
  const float vv  = ws[F2Q + j] * INVN - m * m;
  const float sc2 = ln2w[j] * rsqrtf(vv + EPSV);
  const float sh2 = ln2b[j] - m * sc2;

  const int o0 = 2 * j;
  const float* W3 = allW + ((size_t)b * WTOT + fi + 2 * CMID + o0) * 3;
  const float w0a = W3[0], w1a = W3[1], w2a = W3[2];
  const float w0b = W3[3], w1b = W3[4], w2b = W3[5];

  const int t0 = threadIdx.x * 16;
  const float gml = fmaxf(sc2 * h2r[max(t0 - 1, 0)] + sh2, 0.0f);
  float gm = (t0 > 0) ? gml : 0.0f;
  float gc = fmaxf(sc2 * h2r[t0] + sh2, 0.0f);
  float sa = 0.f, qa = 0.f, sb = 0.f, qb = 0.f;
#pragma unroll 4
  for (int t = t0; t < t0 + 16; ++t) {
    const float gpl = fmaxf(sc2 * h2r[min(t + 1, L2 - 1)] + sh2, 0.0f);
    const float gp = (t + 1 < L2) ? gpl : 0.0f;
    const float va = w0a * gm + w1a * gc + w2a * gp;
    const float vb = w0b * gm + w1b * gc + w2b * gp;
    sa += va; qa += va * va; sb += vb; qb += vb * vb;
    gm = gc; gc = gp;
  }

  __shared__ float4 red[256];
  red[threadIdx.x] = make_float4(sa, qa, sb, qb);
  __syncthreads();
  for (int off = 128; off > 0; off >>= 1) {
    if (threadIdx.x < off) {
      float4 u = red[threadIdx.x], v = red[threadIdx.x + off];
      red[threadIdx.x] = make_float4(u.x + v.x, u.y + v.y, u.z + v.z, u.w + v.w);
    }
    __syncthreads();
  }
  if (threadIdx.x == 0) {
    float4 r = red[0];
    ws[P3S + (size_t)o0 * 32 + b] = r.x;
    ws[P3Q + (size_t)o0 * 32 + b] = r.y;
    ws[P3S + (size_t)(o0 + 1) * 32 + b] = r.z;
    ws[P3Q + (size_t)(o0 + 1) * 32 + b] = r.w;
  }
}

// ---- kernel 3: shortcut GEMM via V_WMMA_F32_16X16X4_F32 + fused epilogue ----
// Block = 512 threads = 16 waves; one block per (b, l-tile of 16).
// Wave w owns the 16-row output tile o in [16w, 16w+16).
__global__ void __launch_bounds__(512)
k3_final(const float* __restrict__ x, const float* __restrict__ allW,
         const float* __restrict__ sw, const float* __restrict__ ln2w,
         const float* __restrict__ ln2b, const float* __restrict__ ln3w,
         const float* __restrict__ ln3b, const int* __restrict__ fip,
         const float* __restrict__ ws, float* __restrict__ out) {
  __shared__ float ldssc[CIN * 16];          // pooled x tile: [c][l], 4 KiB
  const int fi = fip[0];
  const int b  = blockIdx.x >> 8;
  const int lt = blockIdx.x & 255;
  const int l0 = lt * 16;
  const int tid = threadIdx.x;

  // cooperative avg-pool of x into LDS: 1024 values, 2 per thread (float4 read)
  {
    const int idx = tid * 2;
    const int c = idx >> 4, ll = idx & 15;
    const float4 v = *(const float4*)(x + ((size_t)b * CIN + c) * LIN + 2 * (l0 + ll));
    ldssc[idx]     = 0.5f * (v.x + v.y);
    ldssc[idx + 1] = 0.5f * (v.z + v.w);
  }
  __syncthreads();

  const int wave = tid >> 5;
  const int lane = tid & 31;
  const int half = lane >> 4;                // K-group / M-group selector
  const int mrow = lane & 15;                // M for A, N for B/C/D
  const int o0 = wave * 16;
  const float* __restrict__ swrow = sw + (size_t)(o0 + mrow) * CIN;

  // shortcut GEMM: acc(16x16) = shortcut_w[o0:o0+16, :] @ pooled[:, l0:l0+16]
  v8f acc = {};
#pragma unroll
  for (int kk = 0; kk < 16; ++kk) {
    const int cA = kk * 4 + 2 * half;        // this lane supplies K = cA, cA+1
    const float2 av = *(const float2*)(swrow + cA);   // 8B aligned (cA even)
    v2f a, bb;
    a.x  = av.x;
    a.y  = av.y;
    bb.x = ldssc[cA * 16 + mrow];
    bb.y = ldssc[(cA + 1) * 16 + mrow];
    acc = __builtin_amdgcn_wmma_f32_16x16x4_f32(
        /*neg_a=*/false, a, /*neg_b=*/false, bb,
        /*c_mod=*/(short)0, acc, /*reuse_a=*/false, /*reuse_b=*/false);
  }

  const int l = l0 + mrow;
  const int lm = max(l - 1, 0);
  const int lp = min(l + 1, L2 - 1);
  // g = relu(bn2(h2)) triplets for the 4 mid channels this lane's rows need
  float gm[4], gc[4], gp[4];
#pragma unroll
  for (int jj = 0; jj < 4; ++jj) {
    const int ch = (o0 >> 1) + 4 * half + jj;
    const float m   = ws[F2S + ch] * INVN;
    const float vv  = ws[F2Q + ch] * INVN - m * m;
    const float s2  = ln2w[ch] * rsqrtf(vv + EPSV);
    const float sh2 = ln2b[ch] - m * s2;
    const float* __restrict__ h2r = ws + H2O + ((size_t)b * CMID + ch) * L2;
    const float hm = h2r[lm], hc = h2r[l], hp = h2r[lp];   // unconditional loads
    gm[jj] = (l >= 1)     ? fmaxf(s2 * hm + sh2, 0.0f) : 0.0f;
    gc[jj] =                fmaxf(s2 * hc + sh2, 0.0f);
    gp[jj] = (l + 1 < L2) ? fmaxf(s2 * hp + sh2, 0.0f) : 0.0f;
  }

  // epilogue on the WMMA accumulator: D layout -> M = r + 8*half, N = mrow
#pragma unroll
  for (int r = 0; r < 8; ++r) {
    const int o  = o0 + 8 * half + r;
    const int jj = r >> 1;
    const float* W3 = allW + ((size_t)b * WTOT + fi + 2 * CMID + o) * 3;
    const float h3 = W3[0] * gm[jj] + W3[1] * gc[jj] + W3[2] * gp[jj];
    const float m3  = ws[F3S + o] * INVN;
    const float v3  = ws[F3Q + o] * INVN - m3 * m3;
    const float s3  = ln3w[o] * rsqrtf(v3 + EPSV);
    const float sh3 = ln3b[o] - m3 * s3;
    const float val = fmaxf(h3 * s3 + sh3, 0.0f) + acc[r];
    out[((size_t)b * COUT + o) * L2 + l] = fmaxf(val, 0.0f);
  }
}

extern "C" void kernel_launch(void* const* d_in, const int* in_sizes, int n_in,
                              void* d_out, int out_size, void* d_ws, size_t ws_size,
                              hipStream_t stream) {
  const float* x    = (const float*)d_in[0];
  const float* allW = (const float*)d_in[1];
  const float* allb = (const float*)d_in[2];
  const float* sw   = (const float*)d_in[3];
  const float* ln2w = (const float*)d_in[4];
  const float* ln2b = (const float*)d_in[5];
  const float* ln3w = (const float*)d_in[6];
  const float* ln3b = (const float*)d_in[7];
  const int*   fip  = (const int*)d_in[8];
  float* out = (float*)d_out;
  float* ws  = (float*)d_ws;

  k0_tail<<<1, 64, 0, stream>>>(out, fip, out_size);
  k1_conv12<<<B_N * 64, 256, 0, stream>>>(x, allW, allb, fip, ws);
  k_red32<<<1, 128, 0, stream>>>(ws + P2S, ws + P2Q, ws + F2S, ws + F2Q, CMID);
  k2_conv3_stats<<<B_N * CMID, 256, 0, stream>>>(allW, ln2w, ln2b, fip, ws);
  k_red32<<<1, 256, 0, stream>>>(ws + P3S, ws + P3Q, ws + F3S, ws + F3Q, COUT);
  k3_final<<<B_N * 256, 512, 0, stream>>>(x, allW, sw, ln2w, ln2b, ln3w, ln3b,
                                          fip, ws, out);
}